// LIFNeuron_82171314307228
// MI455X (gfx1250) — compile-verified
//
#include <hip/hip_runtime.h>
#include <math.h>

// MI455X (gfx1250) LIF forward.
// N = 33.5M fp32 elements, T = 8 steps. 12 B HBM traffic / elem, ~32 flops/elem
// -> memory bound (AI ~2.7 flop/B). Optimal kernel = NT b128 streaming loads/
// stores + fully unrolled recurrence. Spike-count reduction kept in integers
// (exact + deterministic), per-block partials (no float atomics), finalized by
// a single block that uses v_wmma_f32_16x16x4_f32 (ones-matrix trick) for the
// last 32-lane reduction.

typedef __attribute__((ext_vector_type(4))) float v4f;
typedef __attribute__((ext_vector_type(2))) float v2f;
typedef __attribute__((ext_vector_type(8))) float v8f;

#ifndef THREADS
#define THREADS 256
#endif
#define ITEMS 4  // float4s per thread in the main kernel

__device__ __forceinline__ float dev_sigmoid(float v) {
    return 1.0f / (1.0f + expf(-v));
}

__global__ __launch_bounds__(THREADS) void lif_main_kernel(
    const float* __restrict__ x,
    const float* __restrict__ thr_p,
    const float* __restrict__ tau_p,
    const int*   __restrict__ ts_p,
    float* __restrict__ ssum_out,
    float* __restrict__ mem_out,
    unsigned* __restrict__ partials,
    int n, int n4)
{
    const float thr = thr_p[0];
    const float tau = dev_sigmoid(tau_p[0]);
    const int   T   = ts_p[0];
    const float inv = 1.0f / (float)T;   // exact for T = 8

    const v4f* __restrict__ x4 = (const v4f*)x;
    v4f* __restrict__ s4 = (v4f*)ssum_out;
    v4f* __restrict__ m4 = (v4f*)mem_out;

    unsigned cnt = 0;

    const int base = blockIdx.x * (THREADS * ITEMS) + threadIdx.x;
#pragma unroll
    for (int k = 0; k < ITEMS; ++k) {
        const int i = base + k * THREADS;
        if (i < n4) {
            const v4f xv = __builtin_nontemporal_load(&x4[i]);
            float cu[4] = { xv.x * inv, xv.y * inv, xv.z * inv, xv.w * inv };
            float mm[4] = { 0.0f, 0.0f, 0.0f, 0.0f };
            float sv[4] = { 0.0f, 0.0f, 0.0f, 0.0f };
#pragma unroll 8
            for (int t = 0; t < T; ++t) {
#pragma unroll
                for (int c = 0; c < 4; ++c) {
                    mm[c] = __builtin_fmaf(tau, mm[c], cu[c]);     // m = tau*m + cur
                    const float sp = (mm[c] >= thr) ? 1.0f : 0.0f; // heaviside
                    mm[c] = __builtin_fmaf(sp, -thr, mm[c]);       // soft reset
                    sv[c] += sp;
                }
            }
            v4f so; so.x = sv[0]; so.y = sv[1]; so.z = sv[2]; so.w = sv[3];
            v4f mo; mo.x = mm[0]; mo.y = mm[1]; mo.z = mm[2]; mo.w = mm[3];
            __builtin_nontemporal_store(so, &s4[i]);
            __builtin_nontemporal_store(mo, &m4[i]);
            cnt += (unsigned)(sv[0] + sv[1] + sv[2] + sv[3]);      // exact, <= 32
        }
    }

    // Scalar tail (n not divisible by 4): first (n - 4*n4) global threads.
    {
        const int g    = blockIdx.x * THREADS + threadIdx.x;
        const int tail = n - n4 * 4;
        if (g < tail) {
            const int idx = n4 * 4 + g;
            const float cu = x[idx] * inv;
            float m = 0.0f, ss = 0.0f;
#pragma unroll 8
            for (int t = 0; t < T; ++t) {
                m = __builtin_fmaf(tau, m, cu);
                const float sp = (m >= thr) ? 1.0f : 0.0f;
                m = __builtin_fmaf(sp, -thr, m);
                ss += sp;
            }
            ssum_out[idx] = ss;
            mem_out[idx]  = m;
            cnt += (unsigned)ss;
        }
    }

    // Deterministic integer block reduction -> one partial per block.
    __shared__ unsigned red[THREADS];
    red[threadIdx.x] = cnt;
    __syncthreads();
#pragma unroll
    for (int s = THREADS / 2; s > 0; s >>= 1) {
        if (threadIdx.x < (unsigned)s) red[threadIdx.x] += red[threadIdx.x + s];
        __syncthreads();
    }
    if (threadIdx.x == 0) partials[blockIdx.x] = red[0];
}

__global__ __launch_bounds__(THREADS) void lif_finalize_kernel(
    const unsigned* __restrict__ partials, int nb,
    const int* __restrict__ ts_p,
    float* __restrict__ loss_out,
    long long n)
{
    __shared__ unsigned sm[THREADS];
    unsigned s = 0;
    for (int i = threadIdx.x; i < nb; i += THREADS) s += partials[i];
    sm[threadIdx.x] = s;
    __syncthreads();

    // Fold 256 -> 32 values (every wave computes the identical 32 partials so
    // EXEC is all-ones in every wave for the WMMA below).
    const int lane = threadIdx.x & 31;
    unsigned v = 0;
#pragma unroll
    for (int j = 0; j < 8; ++j) v += sm[lane + 32 * j];
    const float f = (float)v;

    // Final 32-lane reduction via v_wmma_f32_16x16x4_f32 with B = ones:
    // A[r][0] = p_r (lanes 0-15), A[r][2] = p_{r+16} (lanes 16-31), rest 0.
    // D[r][j] = p_r + p_{r+16}.  Sum of a lane's 8 D regs covers rows 0-7
    // (lanes < 16) or rows 8-15 (lanes >= 16); shfl_xor(16) completes the sum.
    v2f a; a.x = f;    a.y = 0.0f;
    v2f b; b.x = 1.0f; b.y = 1.0f;
    v8f c = {0.f, 0.f, 0.f, 0.f, 0.f, 0.f, 0.f, 0.f};
    v8f d = __builtin_amdgcn_wmma_f32_16x16x4_f32(
        /*neg_a=*/false, a, /*neg_b=*/false, b,
        /*c_mod=*/(short)0, c, /*reuse_a=*/false, /*reuse_b=*/false);
    float r = d[0] + d[1] + d[2] + d[3] + d[4] + d[5] + d[6] + d[7];
    const float total = r + __shfl_xor(r, 16, 32);

    if (threadIdx.x == 0) {
        const int T = ts_p[0];
        const double fr   = (double)total / ((double)n * (double)T);
        const double diff = fr - 0.5;
        loss_out[0] = (float)(0.01 * diff * diff);
    }
}

extern "C" void kernel_launch(void* const* d_in, const int* in_sizes, int n_in,
                              void* d_out, int out_size, void* d_ws, size_t ws_size,
                              hipStream_t stream) {
    const float* x     = (const float*)d_in[0];
    const float* thr_p = (const float*)d_in[1];
    const float* tau_p = (const float*)d_in[2];
    const int*   ts_p  = (const int*)d_in[3];

    const int n = in_sizes[0];                  // 33,554,432
    float* out  = (float*)d_out;
    float* ssum = out;                          // [0, n)
    float* mem  = out + (size_t)n;              // [n, 2n)
    float* loss = out + 2 * (size_t)n;          // [2n]
    unsigned* partials = (unsigned*)d_ws;

    const int n4        = n / 4;
    const int per_block = THREADS * ITEMS;      // 1024 float4s / block
    const int tail      = n - 4 * n4;

    int nb_vec  = (n4 + per_block - 1) / per_block;
    int nb_tail = (tail + THREADS - 1) / THREADS;
    int nb = nb_vec > nb_tail ? nb_vec : nb_tail;
    if (nb < 1) nb = 1;                         // 8192 blocks for the given shape

    lif_main_kernel<<<nb, THREADS, 0, stream>>>(
        x, thr_p, tau_p, ts_p, ssum, mem, partials, n, n4);
    lif_finalize_kernel<<<1, THREADS, 0, stream>>>(
        partials, nb, ts_p, loss, (long long)n);
}